// AdditiveAttention_86208583565520
// MI455X (gfx1250) — compile-verified
//
#include <hip/hip_runtime.h>

// ---------------------------------------------------------------------------
// AdditiveAttention for MI455X (gfx1250, wave32)
//   qp  = query@Wq                      (f32 WMMA 16x16x4, LDS-staged A tile)
//   kpT = (key@Wk) stored [b][h][k]     (transposed store -> coalesced h-stream)
//   score[b,q,k] = sum_h wv[h]*tanh(qp+kpT)   (v_tanh_f32, TRANS-pipe bound)
//   attn = masked softmax(score)
//   out  = attn @ value                 (f32 WMMA, LDS-staged attn tile)
// Intermediates (9 MB) live in the 192 MB L2.
// ---------------------------------------------------------------------------

typedef float v2f __attribute__((ext_vector_type(2)));
typedef float v4f __attribute__((ext_vector_type(4)));
typedef float v8f __attribute__((ext_vector_type(8)));
typedef int   v4i __attribute__((__vector_size__(4 * sizeof(int))));

constexpr int B  = 4;
constexpr int LQ = 256;
constexpr int LK = 1024;
constexpr int D  = 512;   // Dq == Dk == Dv
constexpr int H  = 256;
constexpr int DV = 512;

// ---- fast tanh: guaranteed v_tanh_f32 -------------------------------------
__device__ __forceinline__ float fast_tanh(float x) {
#if __has_builtin(__builtin_amdgcn_tanhf)
  return __builtin_amdgcn_tanhf(x);
#else
  float r;
  asm("v_tanh_f32 %0, %1" : "=v"(r) : "v"(x));
  return r;
#endif
}

// ---- async global->LDS copy (16B per lane), ASYNCcnt-tracked ---------------
#if defined(__gfx1250__) && __has_builtin(__builtin_amdgcn_global_load_async_to_lds_b128)
#define HAVE_ASYNC_LDS 1
#endif

#if HAVE_ASYNC_LDS
typedef __attribute__((address_space(1))) v4i* g_v4i_ptr;
typedef __attribute__((address_space(3))) v4i* l_v4i_ptr;
#endif

__device__ __forceinline__ void copy16_g2l(const float* g, float* l) {
#if HAVE_ASYNC_LDS
  void* gnc = (void*)g;                               // drop const (generic AS)
  __builtin_amdgcn_global_load_async_to_lds_b128((g_v4i_ptr)gnc,
                                                 (l_v4i_ptr)l, 0, 0);
#else
  *(v4f*)l = *(const v4f*)g;
#endif
}

__device__ __forceinline__ void wait_async_lds() {
#if HAVE_ASYNC_LDS
#if __has_builtin(__builtin_amdgcn_s_wait_asynccnt)
  __builtin_amdgcn_s_wait_asynccnt(0);
#else
  asm volatile("s_wait_asynccnt 0x0" ::: "memory");
#endif
#endif
}

// ---------------------------------------------------------------------------
// Projection GEMM: C[M,H] = A[M,D] @ W[D,H], fp32 WMMA 16x16x4.
// Block = 8 waves sharing one 16-row A tile (16x512, contiguous) in LDS.
// Pad 516 -> fragment rows land in 16 distinct bank groups (conflict-free).
// transOut=1 stores per-batch transposed:  kpT[b][h][k]
// ---------------------------------------------------------------------------
__global__ void __launch_bounds__(256)
proj_gemm_f32(const float* __restrict__ A, const float* __restrict__ W,
              float* __restrict__ Cout, int transOut) {
  __shared__ float as_[16][516];                      // 33 KB

  const int t    = threadIdx.x;
  const int wid  = blockIdx.x * 8 + (t >> 5);
  const int lane = t & 31;
  const int half = lane >> 4;
  const int l16  = lane & 15;
  const int mt = wid >> 4;                            // 16 n-tiles per m-row
  const int nt = wid & 15;
  const int m0 = mt * 16, n0 = nt * 16;
  const int m0blk = (blockIdx.x >> 1) * 16;           // block-uniform A tile row

  // cooperative stage of A tile rows m0blk..m0blk+15 (contiguous 32 KB)
#pragma unroll
  for (int i = 0; i < 8; ++i) {
    const int chunk = t + 256 * i;                    // 2048 x 16B
    const int row = chunk >> 7;
    const int col = (chunk & 127) * 4;
    copy16_g2l(A + (size_t)(m0blk + row) * D + col, &as_[row][col]);
  }
  wait_async_lds();
  __syncthreads();

  const float* wcol = W + n0 + l16;
  v8f c = {0.f, 0.f, 0.f, 0.f, 0.f, 0.f, 0.f, 0.f};

#pragma unroll 4
  for (int kb = 0; kb < D; kb += 4) {
    const int k2 = kb + 2 * half;
    v2f a = *(const v2f*)&as_[l16][k2];               // ds_read_b64, no conflicts
    v2f b;
    b.x = wcol[(size_t)k2 * H];                       // coalesced 64B per half
    b.y = wcol[(size_t)(k2 + 1) * H];
    c = __builtin_amdgcn_wmma_f32_16x16x4_f32(false, a, false, b,
                                              (short)0, c, false, false);
  }

  if (!transOut) {
#pragma unroll
    for (int r = 0; r < 8; ++r) {
      const int m = m0 + r + 8 * half;
      Cout[(size_t)m * H + n0 + l16] = c[r];
    }
  } else {
#pragma unroll
    for (int r = 0; r < 8; ++r) {
      const int m  = m0 + r + 8 * half;               // global key row
      const int bb = m >> 10;                         // / LK
      const int kk = m & (LK - 1);
      Cout[(size_t)bb * H * LK + (size_t)(n0 + l16) * LK + kk] = c[r];
    }
  }
}

// ---------------------------------------------------------------------------
// Score kernel: scores[b, q0..q0+15, k0..k0+63]
//   per thread: 1 key column, 4 query rows -> 4 tanh per {1 global + 2 LDS} read
//   kpT[b][h][k]: 128B fully-coalesced per wave per h (L2 hit)
//   qs/wvs reads are wave-uniform broadcasts (conflict-free)
// ---------------------------------------------------------------------------
__global__ void __launch_bounds__(256)
scores_tanh(const float* __restrict__ qp, const float* __restrict__ kpT,
            const float* __restrict__ wv, float* __restrict__ scores) {
  __shared__ float qs[16][H];                         // 16 KB (tile contiguous)
  __shared__ float wvs[H];                            // 1 KB

  const int t  = threadIdx.x;
  const int k0 = blockIdx.x * 64;
  const int q0 = blockIdx.y * 16;
  const int b  = blockIdx.z;

  const float* qsrc = qp + ((size_t)b * LQ + q0) * H; // 4096 contiguous floats
#pragma unroll
  for (int i = 0; i < 4; ++i) {
    const int c4 = (t + 256 * i) * 4;
    copy16_g2l(qsrc + c4, &qs[0][0] + c4);
  }
  if (t < 64) copy16_g2l(wv + t * 4, wvs + t * 4);
  wait_async_lds();
  __syncthreads();

  const int k  = k0 + (t & 63);
  const int qr = t >> 6;                              // 0..3 (rows qr+4j)
  const float* kcol = kpT + (size_t)b * H * LK + k;

  float a0 = 0.f, a1 = 0.f, a2 = 0.f, a3 = 0.f;
#pragma unroll 4
  for (int h = 0; h < H; ++h) {
    const float kv = kcol[(size_t)h * LK];
    const float w  = wvs[h];
    a0 = fmaf(w, fast_tanh(qs[qr     ][h] + kv), a0);
    a1 = fmaf(w, fast_tanh(qs[qr +  4][h] + kv), a1);
    a2 = fmaf(w, fast_tanh(qs[qr +  8][h] + kv), a2);
    a3 = fmaf(w, fast_tanh(qs[qr + 12][h] + kv), a3);
  }

  scores[((size_t)b * LQ + q0 + qr     ) * LK + k] = a0;
  scores[((size_t)b * LQ + q0 + qr +  4) * LK + k] = a1;
  scores[((size_t)b * LQ + q0 + qr +  8) * LK + k] = a2;
  scores[((size_t)b * LQ + q0 + qr + 12) * LK + k] = a3;
}

// ---------------------------------------------------------------------------
// Masked softmax over k (row = b*LQ+q, 1024 cols, 4 per thread), in place.
// ---------------------------------------------------------------------------
__global__ void __launch_bounds__(256)
masked_softmax(float* __restrict__ scores, const int* __restrict__ valid_len) {
  __shared__ float red[256];
  const int row  = blockIdx.x;
  const int b    = row >> 8;                          // / LQ
  const int t    = threadIdx.x;
  const int vlen = valid_len[b];
  float* base = scores + (size_t)row * LK;

  float v[4];
  float m = -3.4e38f;
#pragma unroll
  for (int j = 0; j < 4; ++j) {
    const int kidx = t + 256 * j;
    float x = base[kidx];
    x = (kidx < vlen) ? x : -1000000.0f;              // NEG, as in reference
    v[j] = x;
    m = fmaxf(m, x);
  }
  red[t] = m;
  __syncthreads();
  for (int s = 128; s > 0; s >>= 1) {
    if (t < s) red[t] = fmaxf(red[t], red[t + s]);
    __syncthreads();
  }
  const float rmax = red[0];
  __syncthreads();

  float e[4];
  float sum = 0.f;
#pragma unroll
  for (int j = 0; j < 4; ++j) {
    e[j] = __expf(v[j] - rmax);
    sum += e[j];
  }
  red[t] = sum;
  __syncthreads();
  for (int s = 128; s > 0; s >>= 1) {
    if (t < s) red[t] += red[t + s];
    __syncthreads();
  }
  const float inv = 1.0f / red[0];
#pragma unroll
  for (int j = 0; j < 4; ++j) base[t + 256 * j] = e[j] * inv;
}

// ---------------------------------------------------------------------------
// out[b][256][512] = attn[b][256][1024] @ value[b][1024][512], fp32 WMMA.
// Block = 8 waves sharing one 16-row attn tile, staged in 128-col chunks.
// ---------------------------------------------------------------------------
__global__ void __launch_bounds__(256)
attn_value_gemm(const float* __restrict__ attn, const float* __restrict__ V,
                float* __restrict__ out) {
  __shared__ float as_[16][132];                      // 8.25 KB, pad%64==4

  const int t    = threadIdx.x;
  const int wid  = blockIdx.x * 8 + (t >> 5);
  const int lane = t & 31;
  const int half = lane >> 4;
  const int l16  = lane & 15;
  const int b    = wid >> 9;                          // 512 tiles per batch
  const int rem  = wid & 511;
  const int mt = rem >> 5, nt = rem & 31;             // mt,b block-uniform
  const int m0 = mt * 16, n0 = nt * 16;

  const float* arows = attn + ((size_t)b * LQ + m0) * LK;   // block-uniform
  const float* vcol  = V + (size_t)b * LK * DV + n0 + l16;

  v8f c = {0.f, 0.f, 0.f, 0.f, 0.f, 0.f, 0.f, 0.f};

  for (int kc = 0; kc < LK; kc += 128) {
    // stage attn[m0..m0+15][kc..kc+127]
#pragma unroll
    for (int i = 0; i < 2; ++i) {
      const int chunk = t + 256 * i;                  // 512 x 16B
      const int row = chunk >> 5;
      const int col = (chunk & 31) * 4;
      copy16_g2l(arows + (size_t)row * LK + kc + col, &as_[row][col]);
    }
    wait_async_lds();
    __syncthreads();

#pragma unroll 4
    for (int kb = 0; kb < 128; kb += 4) {
      const int k2 = kb + 2 * half;
      v2f a = *(const v2f*)&as_[l16][k2];
      v2f bv;
      bv.x = vcol[(size_t)(kc + k2) * DV];
      bv.y = vcol[(size_t)(kc + k2 + 1) * DV];
      c = __builtin_amdgcn_wmma_f32_16x16x4_f32(false, a, false, bv,
                                                (short)0, c, false, false);
    }
    __syncthreads();
  }

#pragma unroll
  for (int r = 0; r < 8; ++r) {
    const int m = m0 + r + 8 * half;
    out[((size_t)b * LQ + m) * DV + n0 + l16] = c[r];
  }
}

// ---------------------------------------------------------------------------
extern "C" void kernel_launch(void* const* d_in, const int* in_sizes, int n_in,
                              void* d_out, int out_size, void* d_ws,
                              size_t ws_size, hipStream_t stream) {
  (void)in_sizes; (void)n_in; (void)out_size; (void)ws_size;

  const float* query     = (const float*)d_in[0];   // [4,256,512]
  const float* key       = (const float*)d_in[1];   // [4,1024,512]
  const float* value     = (const float*)d_in[2];   // [4,1024,512]
  const int*   valid_len = (const int*)d_in[3];     // [4]
  const float* Wq        = (const float*)d_in[4];   // [512,256]
  const float* Wk        = (const float*)d_in[5];   // [512,256]
  const float* wv        = (const float*)d_in[6];   // [256]
  float* out = (float*)d_out;                       // [4,256,512]

  float* ws     = (float*)d_ws;
  float* qp     = ws;                               // [4,256,256]   1 MB
  float* kpT    = qp + (size_t)B * LQ * H;          // [4,256,1024]  4 MB (T)
  float* scores = kpT + (size_t)B * H * LK;         // [4,256,1024]  4 MB

  proj_gemm_f32<<<128, 256, 0, stream>>>(query, Wq, qp, 0);
  proj_gemm_f32<<<512, 256, 0, stream>>>(key, Wk, kpT, 1);
  scores_tanh<<<dim3(LK / 64, LQ / 16, B), 256, 0, stream>>>(qp, kpT, wv, scores);
  masked_softmax<<<B * LQ, 256, 0, stream>>>(scores, valid_len);
  attn_value_gemm<<<256, 256, 0, stream>>>(scores, value, out);
}